// SoftDecisionTree_936302871357
// MI455X (gfx1250) — compile-verified
//
#include <hip/hip_runtime.h>
#include <hip/hip_bf16.h>

typedef __attribute__((ext_vector_type(2))) float v2f;
typedef __attribute__((ext_vector_type(4))) float v4f;
typedef __attribute__((ext_vector_type(8))) float v8f;

#define BATCH 16384
#define DIM   2048
#define NOUT  1000
#define NIN   255
#define NLEAF 256
#define LQSTR 1024   // padded logQ/expQ row stride
#define KC    16     // K chunk (double-buffered)
#define LSTR  20     // LDS row stride in floats (20*4=80, multiple of 16)

// -------- async global->LDS support (guarded; falls back to sync staging) ----
#ifndef __has_builtin
#define __has_builtin(x) 0
#endif
#if __has_builtin(__builtin_amdgcn_global_load_async_to_lds_b128)
#define HAVE_ASYNC_LDS 1
typedef int v4i __attribute__((vector_size(16)));
typedef __attribute__((address_space(1))) v4i as1_v4i;   // global-space int4
typedef __attribute__((address_space(3))) v4i as3_v4i;   // LDS-space int4
#else
#define HAVE_ASYNC_LDS 0
#endif

__device__ __forceinline__ void wait_async_lds() {
#if HAVE_ASYNC_LDS
#if __has_builtin(__builtin_amdgcn_s_wait_asynccnt)
    __builtin_amdgcn_s_wait_asynccnt(0);
#else
    asm volatile("s_wait_asynccnt 0x0" ::: "memory");
#endif
#endif
}

// ---------------------------------------------------------------------------
// init: zero global accumulators
// ---------------------------------------------------------------------------
__global__ void init_kernel(float* gnum, float* gden, float* gloss) {
    int t = threadIdx.x;
    if (t < 128) gnum[t] = 0.f;
    else gden[t - 128] = 0.f;
    if (t == 0) gloss[0] = 0.f;
}

// ---------------------------------------------------------------------------
// log_softmax over leaf_params rows: logQ (padded stride 1024) and expQ
// ---------------------------------------------------------------------------
__global__ __launch_bounds__(256) void logsoftmax_kernel(
    const float* __restrict__ lp, float* __restrict__ logQ, float* __restrict__ expQ) {
    __shared__ float red[256];
    const int r = blockIdx.x, tid = threadIdx.x;
    const float* row = lp + (size_t)r * NOUT;
    float m = -3.4e38f;
    for (int o = tid; o < NOUT; o += 256) m = fmaxf(m, row[o]);
    red[tid] = m; __syncthreads();
    for (int s = 128; s > 0; s >>= 1) { if (tid < s) red[tid] = fmaxf(red[tid], red[tid + s]); __syncthreads(); }
    m = red[0]; __syncthreads();
    float se = 0.f;
    for (int o = tid; o < NOUT; o += 256) se += __expf(row[o] - m);
    red[tid] = se; __syncthreads();
    for (int s = 128; s > 0; s >>= 1) { if (tid < s) red[tid] += red[tid + s]; __syncthreads(); }
    const float lse = m + __logf(red[0]);
    for (int o = tid; o < NOUT; o += 256) {
        float lq = row[o] - lse;
        logQ[(size_t)r * LQSTR + o] = lq;
        expQ[(size_t)r * LQSTR + o] = __expf(lq);
    }
}

// ---------------------------------------------------------------------------
// target[b] = index of the one-hot 1 in y[b,:]
// ---------------------------------------------------------------------------
__global__ __launch_bounds__(256) void target_kernel(
    const float* __restrict__ y, int* __restrict__ targ) {
    __shared__ int red[256];
    const int b = blockIdx.x, tid = threadIdx.x;
    const float* row = y + (size_t)b * NOUT;
    int loc = -1;
    for (int o = tid; o < NOUT; o += 256) if (row[o] > 0.5f) loc = o;
    red[tid] = loc; __syncthreads();
    for (int s = 128; s > 0; s >>= 1) { if (tid < s) red[tid] = max(red[tid], red[tid + s]); __syncthreads(); }
    if (tid == 0) targ[b] = max(red[0], 0);
}

// ---------------------------------------------------------------------------
// stage one 16-deep K chunk of x (32 rows) and W (255 rows + zero pad) to LDS
// ---------------------------------------------------------------------------
__device__ __forceinline__ void stage_chunk(
    const float* __restrict__ x, const float* __restrict__ W,
    float* xbuf, float* wbuf, int tid, int m0, int k0) {
#if HAVE_ASYNC_LDS
    if (tid < 128) {                        // x tile: 32x16 = 128 b128
        int r = tid >> 2, c = (tid & 3) << 2;
        __builtin_amdgcn_global_load_async_to_lds_b128(
            (as1_v4i*)(x + (size_t)(m0 + r) * DIM + k0 + c),
            (as3_v4i*)(xbuf + r * LSTR + c), 0, 0);
    }
    #pragma unroll
    for (int j = 0; j < 4; ++j) {           // W tile: 256x16 = 1024 b128
        int i = tid + 256 * j;
        int r = i >> 2, c = (i & 3) << 2;
        if (r < NIN)
            __builtin_amdgcn_global_load_async_to_lds_b128(
                (as1_v4i*)(W + (size_t)r * DIM + k0 + c),
                (as3_v4i*)(wbuf + r * LSTR + c), 0, 0);
    }
#else
    if (tid < 128) {
        int r = tid >> 2, c = (tid & 3) << 2;
        *(v4f*)(xbuf + r * LSTR + c) = *(const v4f*)(x + (size_t)(m0 + r) * DIM + k0 + c);
    }
    #pragma unroll
    for (int j = 0; j < 4; ++j) {
        int i = tid + 256 * j;
        int r = i >> 2, c = (i & 3) << 2;
        if (r < NIN)
            *(v4f*)(wbuf + r * LSTR + c) = *(const v4f*)(W + (size_t)r * DIM + k0 + c);
    }
#endif
}

// ---------------------------------------------------------------------------
// probs = sigmoid(beta * (x @ W^T + b))   via V_WMMA_F32_16X16X4_F32
// block: 256 thr = 8 waves; block tile 32(M) x 256(N); wave tile 32x32
// double-buffered LDS, async DMA prefetch overlapped with WMMA compute
// ---------------------------------------------------------------------------
__global__ __launch_bounds__(256) void gemm_probs_kernel(
    const float* __restrict__ x, const float* __restrict__ W,
    const float* __restrict__ bvec, const float* __restrict__ beta,
    float* __restrict__ probs) {
    __shared__ float xs[2][32 * LSTR];      //  5,120 B
    __shared__ float wsh[2][256 * LSTR];    // 40,960 B
    const int tid  = threadIdx.x;
    const int wave = tid >> 5;
    const int lane = tid & 31;
    const int lr = lane & 15;               // row/col within 16
    const int lh = lane >> 4;               // half select -> K pair
    const int m0 = blockIdx.x * 32;
    const int n0 = wave * 32;

    // zero-pad W row 255 in both buffers (never written by staging)
    if (tid < 32) {
        int bf = tid >> 4, c = tid & 15;
        wsh[bf][255 * LSTR + c] = 0.f;
    }

    v8f acc00 = {}, acc01 = {}, acc10 = {}, acc11 = {};

    stage_chunk(x, W, xs[0], wsh[0], tid, m0, 0);

    for (int i = 0; i < DIM / KC; ++i) {
        const int cur = i & 1;
        wait_async_lds();
        __syncthreads();                    // chunk `cur` resident, prior compute done
        if (i + 1 < DIM / KC)
            stage_chunk(x, W, xs[cur ^ 1], wsh[cur ^ 1], tid, m0, (i + 1) * KC);

        const float* xb = xs[cur];
        const float* wb = wsh[cur];
        #pragma unroll
        for (int kk = 0; kk < KC; kk += 4) {
            const int kb = kk + 2 * lh;     // lanes 0-15: K={kk,kk+1}, lanes 16-31: K={kk+2,kk+3}
            v2f a0 = *(const v2f*)(xb + (lr)      * LSTR + kb);
            v2f a1 = *(const v2f*)(xb + (16 + lr) * LSTR + kb);
            v2f b0 = *(const v2f*)(wb + (n0 + lr)      * LSTR + kb);
            v2f b1 = *(const v2f*)(wb + (n0 + 16 + lr) * LSTR + kb);
            acc00 = __builtin_amdgcn_wmma_f32_16x16x4_f32(false, a0, false, b0, (short)0, acc00, false, false);
            acc01 = __builtin_amdgcn_wmma_f32_16x16x4_f32(false, a0, false, b1, (short)0, acc01, false, false);
            acc10 = __builtin_amdgcn_wmma_f32_16x16x4_f32(false, a1, false, b0, (short)0, acc10, false, false);
            acc11 = __builtin_amdgcn_wmma_f32_16x16x4_f32(false, a1, false, b1, (short)0, acc11, false, false);
        }
        __syncthreads();                    // compute done before buffer reuse
    }

    // epilogue: D layout: VGPR r -> M = msub + r + 8*lh ; N = nsub + lane%16
    v8f accs[4] = { acc00, acc01, acc10, acc11 };
    #pragma unroll
    for (int t = 0; t < 4; ++t) {
        const int ms = t >> 1, ns = t & 1;
        const int n = n0 + ns * 16 + lr;
        const float bb = (n < NIN) ? bvec[n] : 0.f;
        const float bt = (n < NIN) ? beta[n] : 0.f;
        #pragma unroll
        for (int r = 0; r < 8; ++r) {
            const int m = m0 + ms * 16 + r + 8 * lh;
            const float z = accs[t][r] + bb;
            const float p = 1.f / (1.f + __expf(-bt * z));
            probs[(size_t)m * NLEAF + n] = p;
        }
    }
}

// ---------------------------------------------------------------------------
// path kernel: one wave per batch row (8 rows per block)
// ---------------------------------------------------------------------------
__global__ __launch_bounds__(256) void path_kernel(
    const float* __restrict__ probs, const float* __restrict__ logQ,
    const int* __restrict__ targ, float* __restrict__ gnum,
    float* __restrict__ gden, float* __restrict__ gloss, int* __restrict__ best) {
    __shared__ float snum[127];
    __shared__ float sden[127];
    __shared__ float sloss;
    const int tid = threadIdx.x, wave = tid >> 5, lane = tid & 31;
    if (tid < 127) { snum[tid] = 0.f; sden[tid] = 0.f; }
    if (tid == 255) sloss = 0.f;
    __syncthreads();

    const int b = blockIdx.x * 8 + wave;
    const float* pr = probs + (size_t)b * NLEAF;
    const int t = targ[b];
    float lsum = 0.f;
    float bestv = -1.f; int besti = 0;

    for (int q = 0; q < 8; ++q) {
        const int l = lane + 32 * q;
        float path = 1.f;
        #pragma unroll
        for (int d = 1; d <= 8; ++d) {
            const int c = l >> (9 - d);                 // column entering level d
            const int node = (1 << (d - 1)) - 1 + c;
            const float p = pr[node];
            if (d < 8 && ((l & ((1 << (9 - d)) - 1)) == 0)) {
                atomicAdd(&snum[node], p * path);       // ds_add_f32
                atomicAdd(&sden[node], path);
            }
            const int bit = (l >> (8 - d)) & 1;
            path *= bit ? p : (1.f - p);
        }
        lsum += path * logQ[(size_t)l * LQSTR + t];
        if (path > bestv) { bestv = path; besti = l; }
    }

    // wave32 reductions
    #pragma unroll
    for (int off = 16; off > 0; off >>= 1) {
        lsum += __shfl_down(lsum, off, 32);
        float ov = __shfl_down(bestv, off, 32);
        int   oi = __shfl_down(besti, off, 32);
        if (ov > bestv || (ov == bestv && oi < besti)) { bestv = ov; besti = oi; }
    }
    if (lane == 0) {
        atomicAdd(&sloss, lsum);
        best[b] = besti;
    }
    __syncthreads();
    if (tid < 127) {
        atomicAdd(&gnum[tid], snum[tid]);
        atomicAdd(&gden[tid], sden[tid]);
    }
    if (tid == 255) atomicAdd(gloss, sloss);
}

// ---------------------------------------------------------------------------
// finalize: C from alphas, loss = -mean(loss_term) + C
// ---------------------------------------------------------------------------
__global__ __launch_bounds__(128) void finalize_kernel(
    const float* __restrict__ gnum, const float* __restrict__ gden,
    const float* __restrict__ gloss, float* __restrict__ out) {
    __shared__ float red[128];
    const int tid = threadIdx.x;
    float term = 0.f;
    if (tid < 127) {
        const int d = 32 - __clz(tid + 1);          // level of node tid (1..7)
        float den = gden[tid];
        den = (den == 0.f) ? 1e-6f : den;
        float a = gnum[tid] / den;
        a = fminf(fmaxf(a, 1e-6f), 1.f - 1e-6f);
        const float lm = ldexpf(0.1f, -d);          // LMBDA * 2^-d
        term = lm * 0.5f * (__logf(a) + log1pf(-a));
    }
    red[tid] = term; __syncthreads();
    for (int s = 64; s > 0; s >>= 1) { if (tid < s) red[tid] += red[tid + s]; __syncthreads(); }
    if (tid == 0) {
        const float C = -red[0];
        out[0] = -(gloss[0] / (float)BATCH) + C;
    }
}

// ---------------------------------------------------------------------------
// output[b,:] = expQ[best[b],:]
// ---------------------------------------------------------------------------
__global__ __launch_bounds__(256) void output_kernel(
    const int* __restrict__ best, const float* __restrict__ expQ,
    float* __restrict__ out) {
    const int b = blockIdx.x, tid = threadIdx.x;
    const float* row = expQ + (size_t)best[b] * LQSTR;
    float* orow = out + 1 + (size_t)b * NOUT;
    for (int o = tid; o < NOUT; o += 256) orow[o] = row[o];
}

// ---------------------------------------------------------------------------
extern "C" void kernel_launch(void* const* d_in, const int* in_sizes, int n_in,
                              void* d_out, int out_size, void* d_ws, size_t ws_size,
                              hipStream_t stream) {
    const float* x    = (const float*)d_in[0];
    const float* y    = (const float*)d_in[1];
    const float* W    = (const float*)d_in[2];
    const float* bvec = (const float*)d_in[3];
    const float* beta = (const float*)d_in[4];
    const float* lp   = (const float*)d_in[5];
    float* out = (float*)d_out;

    float* probs = (float*)d_ws;                         // 16384*256
    float* logQ  = probs + (size_t)BATCH * NLEAF;        // 256*1024
    float* expQ  = logQ + (size_t)NLEAF * LQSTR;         // 256*1024
    float* gnum  = expQ + (size_t)NLEAF * LQSTR;         // 128
    float* gden  = gnum + 128;                           // 128
    float* gloss = gden + 128;                           // 128 (1 used)
    int*   targ  = (int*)(gloss + 128);                  // 16384
    int*   best  = targ + BATCH;                         // 16384

    init_kernel<<<1, 256, 0, stream>>>(gnum, gden, gloss);
    logsoftmax_kernel<<<NLEAF, 256, 0, stream>>>(lp, logQ, expQ);
    target_kernel<<<BATCH, 256, 0, stream>>>(y, targ);
    gemm_probs_kernel<<<BATCH / 32, 256, 0, stream>>>(x, W, bvec, beta, probs);
    path_kernel<<<BATCH / 8, 256, 0, stream>>>(probs, logQ, targ, gnum, gden, gloss, best);
    finalize_kernel<<<1, 128, 0, stream>>>(gnum, gden, gloss, out);
    output_kernel<<<BATCH, 256, 0, stream>>>(best, expQ, out);
}